// GraphSAGE_76673756168564
// MI455X (gfx1250) — compile-verified
//
#include <hip/hip_runtime.h>
#include <hip/hip_bf16.h>

typedef float v2f __attribute__((ext_vector_type(2)));
typedef float v8f __attribute__((ext_vector_type(8)));
typedef int   v4i __attribute__((ext_vector_type(4)));
typedef __attribute__((address_space(3))) v4i* lds_v4i;

#define NNODES 10000
#define KC 64            // K-chunk staged in LDS per iteration
#define LDSTR 68         // LDS row stride (floats): 68%64=4 -> conflict-free b64 reads
#define WBUF (64 * LDSTR)          // one weight panel
#define DBUF (2 * WBUF)            // one double-buffer slot (both weights)

#if defined(__has_builtin)
#  if __has_builtin(__builtin_amdgcn_global_load_async_to_lds_b128)
#    define HAVE_ASYNC_LDS 1
#  endif
#endif
#ifndef HAVE_ASYNC_LDS
#  define HAVE_ASYNC_LDS 0
#endif

__device__ __forceinline__ void wait_async0() {
#if HAVE_ASYNC_LDS
#  if __has_builtin(__builtin_amdgcn_s_wait_asynccnt)
  __builtin_amdgcn_s_wait_asynccnt(0);
#  else
  asm volatile("s_wait_asynccnt 0x0" ::: "memory");
#  endif
#endif
}

// ---------------- utility kernels ----------------

__global__ __launch_bounds__(256) void fill_zero_f32(float* __restrict__ p, int n) {
  int t = blockIdx.x * blockDim.x + threadIdx.x;
  if (t < n) p[t] = 0.0f;
}

__global__ __launch_bounds__(256) void deg_count(const int* __restrict__ dst,
                                                 float* __restrict__ deg, int E) {
  int t = blockIdx.x * blockDim.x + threadIdx.x;
  if (t < E) unsafeAtomicAdd(&deg[dst[t]], 1.0f);
}

__global__ __launch_bounds__(256) void deg_invert(float* __restrict__ deg, int n) {
  int t = blockIdx.x * blockDim.x + threadIdx.x;
  if (t < n) {
    float d = deg[t];
    deg[t] = (d > 0.0f) ? (1.0f / d) : 0.0f;
  }
}

// stage a [NROWS x KC] chunk of both weight matrices into LDS (async when available)
template <int NROWS>
__device__ __forceinline__ void stage_wchunk(
    const float* __restrict__ Wl, const float* __restrict__ Wr,
    float* __restrict__ dL, float* __restrict__ dR,
    int K, int k0, int lr, int lq)
{
  const bool full = (k0 + KC <= K);
#pragma unroll
  for (int r = lr; r < NROWS; r += 16) {
    const float* gl = Wl + (size_t)r * K + k0 + lq;
    const float* gr = Wr + (size_t)r * K + k0 + lq;
    float* dl = dL + r * LDSTR + lq;
    float* dr = dR + r * LDSTR + lq;
    if (full) {
#if HAVE_ASYNC_LDS
      __builtin_amdgcn_global_load_async_to_lds_b128((v4i*)gl, (lds_v4i)dl, 0, 0);
      __builtin_amdgcn_global_load_async_to_lds_b128((v4i*)gr, (lds_v4i)dr, 0, 0);
#else
      *(float4*)dl = *(const float4*)gl;
      *(float4*)dr = *(const float4*)gr;
#endif
    } else {                     // K tail: zero-pad, synchronous guarded loads
      float4 vl = make_float4(0.f, 0.f, 0.f, 0.f);
      float4 vr = make_float4(0.f, 0.f, 0.f, 0.f);
      int rem = K - (k0 + lq);
      if (rem > 0) { vl.x = gl[0]; vr.x = gr[0]; }
      if (rem > 1) { vl.y = gl[1]; vr.y = gr[1]; }
      if (rem > 2) { vl.z = gl[2]; vr.z = gr[2]; }
      if (rem > 3) { vl.w = gl[3]; vr.w = gr[3]; }
      *(float4*)dl = vl;
      *(float4*)dr = vr;
    }
  }
}

// ---------------- fused dual GEMM: Yl = X@Wl^T, Yr = X@Wr^T ----------------
// Block = 256 threads = 8 waves; each wave owns one 16-row M-tile, full N width.
// Weight K-chunks double-buffered in LDS (async prefetch), shared by all 8 waves.
template <int NT>
__global__ __launch_bounds__(256) void sage_dual_gemm_wmma(
    const float* __restrict__ X, const float* __restrict__ Wl,
    const float* __restrict__ Wr, float* __restrict__ Yl, float* __restrict__ Yr,
    int M, int K)
{
  constexpr int N = NT * 16;
  __shared__ float lw[2 * DBUF];           // [buffer][l/r][row*LDSTR + k]

  const int tid  = threadIdx.x;
  const int wid  = tid >> 5;
  const int lane = tid & 31;
  const int l16  = lane & 15;
  const int koff = (lane >> 4) << 1;       // 0 (lanes 0-15) or 2 (lanes 16-31)

  int row0 = (blockIdx.x * 8 + wid) << 4;
  const bool active = (row0 < M);
  if (row0 > M - 16) row0 = M - 16;        // clamp: inactive waves do redundant work

  const float* __restrict__ xrow = X + (size_t)(row0 + l16) * K;

  v8f accl[NT] = {};
  v8f accr[NT] = {};

  // staging map: 16 consecutive threads cover one weight row's 64-float chunk
  const int lr = tid >> 4;                 // 0..15
  const int lq = (tid & 15) << 2;          // 0,4,...,60

  float* const lbase = &lw[0];
  const int nchunks = (K + KC - 1) / KC;

  stage_wchunk<N>(Wl, Wr, lbase, lbase + WBUF, K, 0, lr, lq);
  wait_async0();
  __syncthreads();

  int p = 0;
  for (int c = 0; c < nchunks; ++c) {
    const int k0 = c * KC;
    if (c + 1 < nchunks) {                 // prefetch next chunk into other buffer
      float* nb = lbase + (p ^ 1) * DBUF;
      stage_wchunk<N>(Wl, Wr, nb, nb + WBUF, K, k0 + KC, lr, lq);
    }

    const float* __restrict__ curL = lbase + p * DBUF;
    const float* __restrict__ curR = curL + WBUF;

    int kvalid = K - k0; if (kvalid > KC) kvalid = KC;
    const int kmain = kvalid & ~3;
    for (int kk = 0; kk < kmain; kk += 4) {
      v2f a;
      a.x = xrow[k0 + kk + koff];
      a.y = xrow[k0 + kk + koff + 1];
#pragma unroll
      for (int n = 0; n < NT; ++n) {
        const int bidx = (n * 16 + l16) * LDSTR + kk + koff;
        v2f bl = *(const v2f*)&curL[bidx];
        v2f br = *(const v2f*)&curR[bidx];
        accl[n] = __builtin_amdgcn_wmma_f32_16x16x4_f32(false, a, false, bl,
                                                        (short)0, accl[n], false, false);
        accr[n] = __builtin_amdgcn_wmma_f32_16x16x4_f32(false, a, false, br,
                                                        (short)0, accr[n], false, false);
      }
    }
    if (kmain < kvalid) {                   // K tail (1..3); LDS zero-padded there
      v2f a = {};
      if (kmain + koff < kvalid)     a.x = xrow[k0 + kmain + koff];
      if (kmain + koff + 1 < kvalid) a.y = xrow[k0 + kmain + koff + 1];
#pragma unroll
      for (int n = 0; n < NT; ++n) {
        const int bidx = (n * 16 + l16) * LDSTR + kmain + koff;
        v2f bl = *(const v2f*)&curL[bidx];
        v2f br = *(const v2f*)&curR[bidx];
        accl[n] = __builtin_amdgcn_wmma_f32_16x16x4_f32(false, a, false, bl,
                                                        (short)0, accl[n], false, false);
        accr[n] = __builtin_amdgcn_wmma_f32_16x16x4_f32(false, a, false, br,
                                                        (short)0, accr[n], false, false);
      }
    }

    wait_async0();                          // ASYNCcnt not covered by the barrier
    __syncthreads();
    p ^= 1;
  }

  if (active) {
    const int rbase = row0 + ((lane >> 4) << 3);
#pragma unroll
    for (int n = 0; n < NT; ++n) {
      const int c = n * 16 + l16;
#pragma unroll
      for (int v = 0; v < 8; ++v) {
        Yl[(size_t)(rbase + v) * N + c] = accl[n][v];
        Yr[(size_t)(rbase + v) * N + c] = accr[n][v];
      }
    }
  }
}

// ---------------- single WMMA GEMM (post_mp): C = X@W^T + bias ----------------
__global__ __launch_bounds__(256) void gemm_xt_wmma(
    const float* __restrict__ X, const float* __restrict__ W,
    const float* __restrict__ bias, float* __restrict__ C,
    int M, int N, int K)
{
  const int wave = blockIdx.x * (blockDim.x >> 5) + (threadIdx.x >> 5);
  const int lane = threadIdx.x & 31;
  const int ntn  = N >> 4;
  const int tm   = wave / ntn;
  const int tn   = wave - tm * ntn;
  if (tm * 16 >= M) return;                 // wave-uniform

  const int row0 = tm * 16, col0 = tn * 16;
  const int l16  = lane & 15;
  const int koff = (lane >> 4) << 1;

  const float* __restrict__ xrow = X + (size_t)(row0 + l16) * K;
  const float* __restrict__ wrow = W + (size_t)(col0 + l16) * K;

  v8f acc = {};
  const int k4 = K & ~3;
  for (int k0 = 0; k0 < k4; k0 += 4) {
    v2f a, b;
    a.x = xrow[k0 + koff];     a.y = xrow[k0 + koff + 1];
    b.x = wrow[k0 + koff];     b.y = wrow[k0 + koff + 1];
    acc = __builtin_amdgcn_wmma_f32_16x16x4_f32(false, a, false, b,
                                                (short)0, acc, false, false);
  }
  if (k4 < K) {
    v2f a = {}, b = {};
    if (k4 + koff < K)     { a.x = xrow[k4 + koff];     b.x = wrow[k4 + koff]; }
    if (k4 + koff + 1 < K) { a.y = xrow[k4 + koff + 1]; b.y = wrow[k4 + koff + 1]; }
    acc = __builtin_amdgcn_wmma_f32_16x16x4_f32(false, a, false, b,
                                                (short)0, acc, false, false);
  }

  float bv = (bias != nullptr) ? bias[col0 + l16] : 0.0f;
  const int rbase = row0 + ((lane >> 4) << 3);
#pragma unroll
  for (int v = 0; v < 8; ++v)
    C[(size_t)(rbase + v) * N + col0 + l16] = acc[v] + bv;
}

// ---------------- edge scatter: agg[dst] += y[src], D multiple of 4 ----------------
__global__ __launch_bounds__(256) void scatter_add_edges(
    const int* __restrict__ src, const int* __restrict__ dst,
    const float* __restrict__ y, float* __restrict__ agg, int E, int D)
{
  int t = blockIdx.x * blockDim.x + threadIdx.x;
  const int chunks = D >> 2;
  if (t >= E * chunks) return;
  int e = t / chunks;
  int c = (t - e * chunks) << 2;
  int s = src[e], d = dst[e];
  float4 v = *(const float4*)(y + (size_t)s * D + c);
  float* ap = agg + (size_t)d * D + c;
  unsafeAtomicAdd(ap + 0, v.x);
  unsafeAtomicAdd(ap + 1, v.y);
  unsafeAtomicAdd(ap + 2, v.z);
  unsafeAtomicAdd(ap + 3, v.w);
}

// ---------------- h = relu(agg * deginv + yr + bias) ----------------
__global__ __launch_bounds__(256) void combine_relu(
    float* __restrict__ h, const float* __restrict__ agg,
    const float* __restrict__ yr, const float* __restrict__ bias,
    const float* __restrict__ deginv, int N, int D)
{
  int t = blockIdx.x * blockDim.x + threadIdx.x;
  if (t >= N * D) return;
  int n = t / D;
  int d = t - n * D;
  float v = agg[t] * deginv[n] + yr[t] + bias[d];
  h[t] = v > 0.0f ? v : 0.0f;
}

// ---------------- head: logits = t1 @ M2w^T + M2b, then log_softmax over 7 ----------------
__global__ __launch_bounds__(256) void head_logsoftmax(
    const float* __restrict__ t1, const float* __restrict__ M2w,
    const float* __restrict__ M2b, float* __restrict__ out, int N)
{
  int n = blockIdx.x * blockDim.x + threadIdx.x;
  if (n >= N) return;
  const float* __restrict__ x = t1 + (size_t)n * 32;
  float logits[7];
#pragma unroll
  for (int c = 0; c < 7; ++c) {
    float s = M2b[c];
    const float* __restrict__ w = M2w + c * 32;
#pragma unroll
    for (int k = 0; k < 32; ++k) s += x[k] * w[k];
    logits[c] = s;
  }
  float mx = logits[0];
#pragma unroll
  for (int c = 1; c < 7; ++c) mx = fmaxf(mx, logits[c]);
  float sum = 0.0f;
#pragma unroll
  for (int c = 0; c < 7; ++c) sum += __expf(logits[c] - mx);
  float lse = mx + __logf(sum);
#pragma unroll
  for (int c = 0; c < 7; ++c) out[(size_t)n * 7 + c] = logits[c] - lse;
}

// ---------------- host side ----------------

static inline int cdiv(int a, int b) { return (a + b - 1) / b; }

extern "C" void kernel_launch(void* const* d_in, const int* in_sizes, int n_in,
                              void* d_out, int out_size, void* d_ws, size_t ws_size,
                              hipStream_t stream) {
  const float* x   = (const float*)d_in[0];
  const int*   ei  = (const int*)d_in[1];
  const float* W1l = (const float*)d_in[2];
  const float* b1  = (const float*)d_in[3];
  const float* W1r = (const float*)d_in[4];
  const float* W2l = (const float*)d_in[5];
  const float* b2  = (const float*)d_in[6];
  const float* W2r = (const float*)d_in[7];
  const float* W3l = (const float*)d_in[8];
  const float* b3  = (const float*)d_in[9];
  const float* W3r = (const float*)d_in[10];
  const float* M1w = (const float*)d_in[11];
  const float* M1b = (const float*)d_in[12];
  const float* M2w = (const float*)d_in[13];
  const float* M2b = (const float*)d_in[14];

  const int N = NNODES;
  const int E = in_sizes[1] / 2;
  const int* src = ei;
  const int* dst = ei + E;

  // workspace layout (floats)
  float* ws   = (float*)d_ws;
  float* deg  = ws;                          // N (becomes deg_inv in place)
  float* bufA = ws + 16384;                  // 640000   yl / yl2 / yl3 / t1
  float* bufB = bufA + 640000;               // 640000   yr / yr2 / yr3
  float* bufC = bufB + 640000;               // 640000   agg per layer
  float* h1   = bufC + 640000;               // 640000   h1, later h3
  float* h2   = h1 + 640000;                 // 320000   h2
  float* h3   = h1;                          // reuse h1 region
  float* t1   = bufA;                        // reuse for post_mp

  const int mblocks = cdiv(N / 16, 8);       // 8 waves (M-tiles) per block

  // degree -> deg_inv
  fill_zero_f32<<<cdiv(N, 256), 256, 0, stream>>>(deg, N);
  deg_count<<<cdiv(E, 256), 256, 0, stream>>>(dst, deg, E);
  deg_invert<<<cdiv(N, 256), 256, 0, stream>>>(deg, N);

  // ---- layer 1: 1433 -> 64 (fused dual GEMM) ----
  sage_dual_gemm_wmma<4><<<mblocks, 256, 0, stream>>>(x, W1l, W1r, bufA, bufB, N, 1433);
  fill_zero_f32<<<cdiv(N * 64, 256), 256, 0, stream>>>(bufC, N * 64);
  scatter_add_edges<<<cdiv(E * 16, 256), 256, 0, stream>>>(src, dst, bufA, bufC, E, 64);
  combine_relu<<<cdiv(N * 64, 256), 256, 0, stream>>>(h1, bufC, bufB, b1, deg, N, 64);

  // ---- layer 2: 64 -> 32 ----
  sage_dual_gemm_wmma<2><<<mblocks, 256, 0, stream>>>(h1, W2l, W2r, bufA, bufB, N, 64);
  fill_zero_f32<<<cdiv(N * 32, 256), 256, 0, stream>>>(bufC, N * 32);
  scatter_add_edges<<<cdiv(E * 8, 256), 256, 0, stream>>>(src, dst, bufA, bufC, E, 32);
  combine_relu<<<cdiv(N * 32, 256), 256, 0, stream>>>(h2, bufC, bufB, b2, deg, N, 32);

  // ---- layer 3: 32 -> 32 ----
  sage_dual_gemm_wmma<2><<<mblocks, 256, 0, stream>>>(h2, W3l, W3r, bufA, bufB, N, 32);
  fill_zero_f32<<<cdiv(N * 32, 256), 256, 0, stream>>>(bufC, N * 32);
  scatter_add_edges<<<cdiv(E * 8, 256), 256, 0, stream>>>(src, dst, bufA, bufC, E, 32);
  combine_relu<<<cdiv(N * 32, 256), 256, 0, stream>>>(h3, bufC, bufB, b3, deg, N, 32);

  // ---- post_mp ----
  {
    int tiles = (N / 16) * (32 / 16);
    gemm_xt_wmma<<<cdiv(tiles, 8), 256, 0, stream>>>(h3, M1w, M1b, t1, N, 32, 32);
  }
  head_logsoftmax<<<cdiv(N, 256), 256, 0, stream>>>(t1, M2w, M2b, (float*)d_out, N);
}